// GATModel_89232240542051
// MI455X (gfx1250) — compile-verified
//
#include <hip/hip_runtime.h>

#define NNODES 50000
#define NEDGES 1600000
#define FDIM   128      // HEADS*HID = HEADS*OUT = 128 everywhere
#define HEADS  4
#define HID    32
#define NEG_SLOPE 0.2f

typedef __attribute__((ext_vector_type(2))) float v2f;
typedef __attribute__((ext_vector_type(8))) float v8f;

// ---------------------------------------------------------------------------
// GEMM: Y[N,128] = X[N,128] @ W[128,128], fp32 WMMA (V_WMMA_F32_16X16X4_F32).
// One wave computes one 16x16 tile; 8 waves/block cover the 128 output cols.
// A(16x4) layout: lanes 0-15 -> M=lane, VGPR0/1 = K=0/1; lanes 16-31 K=2/3.
// B(4x16) layout: lanes 0-15 -> N=lane, VGPR0/1 = K=0/1; lanes 16-31 K=2/3.
// C(16x16) layout: lanes 0-15 N=lane M=r; lanes 16-31 N=lane-16 M=8+r.
// ---------------------------------------------------------------------------
__global__ void gemm_wmma_f32(const float* __restrict__ X,
                              const float* __restrict__ W,
                              float* __restrict__ Y) {
    const int wave = threadIdx.x >> 5;
    const int lane = threadIdx.x & 31;
    const int lh   = lane & 15;      // row (A) / col (B,C) within tile
    const int kg   = lane >> 4;      // K-pair group: 0 -> K{0,1}, 1 -> K{2,3}
    const int row0 = blockIdx.x * 16;
    const int col0 = wave * 16;

    const float* __restrict__ xrow = X + (size_t)(row0 + lh) * FDIM;

    v8f acc = {};
#pragma unroll
    for (int k = 0; k < FDIM; k += 4) {
        v2f a, b;
        a.x = xrow[k + 2 * kg + 0];
        a.y = xrow[k + 2 * kg + 1];
        b.x = W[(size_t)(k + 2 * kg + 0) * FDIM + col0 + lh];
        b.y = W[(size_t)(k + 2 * kg + 1) * FDIM + col0 + lh];
        acc = __builtin_amdgcn_wmma_f32_16x16x4_f32(
            /*neg_a=*/false, a, /*neg_b=*/false, b,
            /*c_mod=*/(short)0, acc, /*reuse_a=*/false, /*reuse_b=*/false);
    }

    const int mbase = (lane < 16) ? 0 : 8;
#pragma unroll
    for (int r = 0; r < 8; ++r)
        Y[(size_t)(row0 + mbase + r) * FDIM + col0 + lh] = acc[r];
}

// ---------------------------------------------------------------------------
// Per-node attention coefficients: el[n,h] = sum_d ft[n,h,d]*al[h,d], er same.
// Block = 128 threads = 4 waves; each wave32 is exactly one head (wave32!).
// ---------------------------------------------------------------------------
__global__ void attn_coef(const float* __restrict__ ft,
                          const float* __restrict__ al,
                          const float* __restrict__ ar,
                          float* __restrict__ el,
                          float* __restrict__ er) {
    const int n = blockIdx.x;
    const int t = threadIdx.x;            // 0..127, h = t>>5, d = t&31
    const float f = ft[(size_t)n * FDIM + t];
    float vl = f * al[t];
    float vr = f * ar[t];
#pragma unroll
    for (int off = 16; off > 0; off >>= 1) {
        vl += __shfl_down(vl, off, 32);
        vr += __shfl_down(vr, off, 32);
    }
    if ((t & 31) == 0) {
        el[n * HEADS + (t >> 5)] = vl;
        er[n * HEADS + (t >> 5)] = vr;
    }
}

__global__ void zero_buf(float* __restrict__ p, int n) {
    int i = blockIdx.x * blockDim.x + threadIdx.x;
    if (i < n) p[i] = 0.f;
}

__global__ void init_mz(float* __restrict__ m, float* __restrict__ z, int n) {
    int i = blockIdx.x * blockDim.x + threadIdx.x;
    if (i < n) { m[i] = -3.402823466e38f; z[i] = 0.f; }
}

__device__ __forceinline__ float leaky(float v) {
    return v > 0.f ? v : NEG_SLOPE * v;
}

// sign-aware float atomic max -> lowers to global_atomic_max_i32/min_u32
__device__ __forceinline__ void atomicMaxF(float* addr, float v) {
    if (v >= 0.f) atomicMax((int*)addr, __float_as_int(v));
    else          atomicMin((unsigned int*)addr, __float_as_uint(v));
}

// Pass 1: per-destination segment max of leaky_relu(el[src]+er[dst])
__global__ void edge_max(const int* __restrict__ src, const int* __restrict__ dst,
                         const float* __restrict__ el, const float* __restrict__ er,
                         float* __restrict__ m) {
    int i = blockIdx.x * blockDim.x + threadIdx.x;
    if (i >= NEDGES * HEADS) return;
    int e = i >> 2, h = i & 3;
    int s = src[e], d = dst[e];
    float v = leaky(el[s * HEADS + h] + er[d * HEADS + h]);
    atomicMaxF(&m[d * HEADS + h], v);
}

// Pass 2: softmax denominator z[dst,h] += exp(e - m[dst,h])
__global__ void edge_sum(const int* __restrict__ src, const int* __restrict__ dst,
                         const float* __restrict__ el, const float* __restrict__ er,
                         const float* __restrict__ m, float* __restrict__ z) {
    int i = blockIdx.x * blockDim.x + threadIdx.x;
    if (i >= NEDGES * HEADS) return;
    int e = i >> 2, h = i & 3;
    int s = src[e], d = dst[e];
    float v = leaky(el[s * HEADS + h] + er[d * HEADS + h]);
    atomicAdd(&z[d * HEADS + h], __expf(v - m[d * HEADS + h]));
}

// Pass 3: rst[dst] += a * ft[src]. 32 lanes per edge; lane owns d=lane for all
// 4 heads -> 128 atomic f32 adds per edge (fully coalesced per wave).
__global__ void edge_aggr(const int* __restrict__ src, const int* __restrict__ dst,
                          const float* __restrict__ ft,
                          const float* __restrict__ el, const float* __restrict__ er,
                          const float* __restrict__ m, const float* __restrict__ z,
                          float* __restrict__ rst) {
    long long gid = (long long)blockIdx.x * blockDim.x + threadIdx.x;
    int e = (int)(gid >> 5);
    int lane = (int)(gid & 31);
    if (e >= NEDGES) return;
    int s = src[e], d = dst[e];
    const float* __restrict__ fs = ft + (size_t)s * FDIM;
    float* __restrict__ rd = rst + (size_t)d * FDIM;
    float a[HEADS];
#pragma unroll
    for (int h = 0; h < HEADS; ++h) {
        float v = leaky(el[s * HEADS + h] + er[d * HEADS + h]);
        a[h] = __expf(v - m[d * HEADS + h]) / z[d * HEADS + h];
    }
#pragma unroll
    for (int h = 0; h < HEADS; ++h)
        atomicAdd(&rd[h * HID + lane], fs[h * HID + lane] * a[h]);
}

// layers 1,2: x_out = elu(rst + b)
__global__ void elu_bias(const float* __restrict__ rst, const float* __restrict__ b,
                         float* __restrict__ xout) {
    int i = blockIdx.x * blockDim.x + threadIdx.x;
    if (i >= NNODES * FDIM) return;
    float v = rst[i] + b[i & (FDIM - 1)];
    xout[i] = v > 0.f ? v : (__expf(v) - 1.f);
}

// layer 3: out[n,d] = mean_h (rst[n,h,d] + b[h,d])
__global__ void mean_heads(const float* __restrict__ rst, const float* __restrict__ b,
                           float* __restrict__ out) {
    int i = blockIdx.x * blockDim.x + threadIdx.x;
    if (i >= NNODES * HID) return;
    int n = i >> 5, dd = i & 31;
    float acc = 0.f;
#pragma unroll
    for (int h = 0; h < HEADS; ++h)
        acc += rst[(size_t)n * FDIM + h * HID + dd] + b[h * HID + dd];
    out[i] = acc * 0.25f;
}

// ---------------------------------------------------------------------------
static void run_layer(const float* x, const float* W, const float* al,
                      const float* ar, const float* b, const int* src,
                      const int* dst, float* ft, float* rst, float* el,
                      float* er, float* m, float* z, float* x_out,
                      bool last, float* out, hipStream_t stream) {
    gemm_wmma_f32<<<NNODES / 16, 256, 0, stream>>>(x, W, ft);
    attn_coef<<<NNODES, FDIM, 0, stream>>>(ft, al, ar, el, er);
    zero_buf<<<(NNODES * FDIM + 255) / 256, 256, 0, stream>>>(rst, NNODES * FDIM);
    init_mz<<<(NNODES * HEADS + 255) / 256, 256, 0, stream>>>(m, z, NNODES * HEADS);
    edge_max<<<(NEDGES * HEADS + 255) / 256, 256, 0, stream>>>(src, dst, el, er, m);
    edge_sum<<<(NEDGES * HEADS + 255) / 256, 256, 0, stream>>>(src, dst, el, er, m, z);
    edge_aggr<<<(NEDGES * 32) / 256, 256, 0, stream>>>(src, dst, ft, el, er, m, z, rst);
    if (last)
        mean_heads<<<(NNODES * HID + 255) / 256, 256, 0, stream>>>(rst, b, out);
    else
        elu_bias<<<(NNODES * FDIM + 255) / 256, 256, 0, stream>>>(rst, b, x_out);
}

extern "C" void kernel_launch(void* const* d_in, const int* in_sizes, int n_in,
                              void* d_out, int out_size, void* d_ws, size_t ws_size,
                              hipStream_t stream) {
    const float* h   = (const float*)d_in[0];
    const int*   src = (const int*)d_in[1];
    const int*   dst = (const int*)d_in[2];
    const float* W1  = (const float*)d_in[3];
    const float* al1 = (const float*)d_in[4];
    const float* ar1 = (const float*)d_in[5];
    const float* b1  = (const float*)d_in[6];
    const float* W2  = (const float*)d_in[7];
    const float* al2 = (const float*)d_in[8];
    const float* ar2 = (const float*)d_in[9];
    const float* b2  = (const float*)d_in[10];
    const float* W3  = (const float*)d_in[11];
    const float* al3 = (const float*)d_in[12];
    const float* ar3 = (const float*)d_in[13];
    const float* b3  = (const float*)d_in[14];
    float* out = (float*)d_out;

    float* ws  = (float*)d_ws;
    float* ft  = ws;                       // N*128
    float* rst = ft + (size_t)NNODES * FDIM;   // N*128
    float* el  = rst + (size_t)NNODES * FDIM;  // N*4
    float* er  = el + NNODES * HEADS;
    float* m   = er + NNODES * HEADS;
    float* z   = m + NNODES * HEADS;

    // layer 1: input h, output (elu) written into ft buffer (ft dead after aggr)
    run_layer(h, W1, al1, ar1, b1, src, dst, ft, rst, el, er, m, z, ft,
              false, nullptr, stream);
    // layer 2: input ft, output back into ft
    run_layer(ft, W2, al2, ar2, b2, src, dst, ft, rst, el, er, m, z, ft,
              false, nullptr, stream);
    // layer 3: head-mean into d_out
    run_layer(ft, W3, al3, ar3, b3, src, dst, ft, rst, el, er, m, z, nullptr,
              true, out, stream);
}